// GeometricPositionalFingerprinter_51213190038177
// MI455X (gfx1250) — compile-verified
//
#include <hip/hip_runtime.h>
#include <hip/hip_bf16.h>
#include <math.h>
#include <stdint.h>

typedef float v2f __attribute__((ext_vector_type(2)));
typedef float v8f __attribute__((ext_vector_type(8)));
typedef int   v4i __attribute__((vector_size(4 * sizeof(int))));
typedef __attribute__((address_space(1))) v4i* g4ptr;   // global int4*
typedef __attribute__((address_space(3))) v4i* l4ptr;   // LDS int4*

#define DIMK        128
#define ROWSTRIDE   132        // 128 + 4 floats pad to break LDS bank alignment
#define WAVES       4          // waves per block (blockDim = 128)
#define PPW         3          // pentachora per wave (15 of 16 tile rows used)

#if __has_builtin(__builtin_amdgcn_global_load_async_to_lds_b128) && \
    __has_builtin(__builtin_amdgcn_s_wait_asynccnt)
#define USE_ASYNC 1
#else
#define USE_ASYNC 0
#endif

__device__ __forceinline__ void copy16_to_lds(const float* g, float* l) {
#if USE_ASYNC
    float* gnc = (float*)g;                 // drop const first
    __builtin_amdgcn_global_load_async_to_lds_b128((g4ptr)gnc, (l4ptr)l, 0, 0);
#else
    *(float4*)l = *(const float4*)g;
#endif
}

__device__ __forceinline__ float sigmoidf_(float x) {
    return 1.0f / (1.0f + expf(-x));
}

__global__ void __launch_bounds__(WAVES * 32)
pentachora_fingerprint_kernel(const float* __restrict__ P, float* __restrict__ out, int V)
{
    __shared__ float tile[WAVES][16][ROWSTRIDE];   // 33,792 B
    __shared__ float gbuf[WAVES][16 * 16];         //  4,096 B

    const int lane = threadIdx.x & 31;
    const int wave = threadIdx.x >> 5;
    const int hl   = lane >> 4;       // which 16-lane half
    const int r    = lane & 15;       // tile row (A) / column (B,C)

    const long long tileId   = (long long)blockIdx.x * WAVES + wave;
    const long long pentBase = tileId * PPW;
    const long long totalRows = (long long)V * 5;

    // ---- Stage 16x128 f32 tile (rows 0..14 = 3 pentachora) into LDS ----
    // zero unused row 15 so the Gram has no wild values
    *(float4*)&tile[wave][15][lane * 4] = make_float4(0.f, 0.f, 0.f, 0.f);

    #pragma unroll
    for (int row = 0; row < 15; ++row) {
        long long grow = pentBase * 5 + row;
        if (grow >= totalRows) grow = totalRows - 1;   // tail clamp; results discarded
        const float* src = P + grow * DIMK + lane * 4;
        float*       dst = &tile[wave][row][lane * 4];
        copy16_to_lds(src, dst);
    }
#if USE_ASYNC
    __builtin_amdgcn_s_wait_asynccnt(0);
#endif
    __asm__ volatile("" ::: "memory");

    // ---- G = X * X^T via V_WMMA_F32_16X16X4_F32, K = 128 (32 steps) ----
    // 32-bit A(16x4) fragment for lane: X[r][4k + 2*hl + {0,1}]
    // 32-bit B(4x16) fragment for lane: X[r][4k + 2*hl + {0,1}]  (identical!)
    v8f acc0 = {0.f, 0.f, 0.f, 0.f, 0.f, 0.f, 0.f, 0.f};
    v8f acc1 = acc0;
    const float* lrow = &tile[wave][r][2 * hl];
    #pragma unroll
    for (int k = 0; k < 32; k += 2) {
        v2f a0 = *(const v2f*)(lrow + 4 * k);
        v2f a1 = *(const v2f*)(lrow + 4 * (k + 1));
        acc0 = __builtin_amdgcn_wmma_f32_16x16x4_f32(false, a0, false, a0,
                                                     (short)0, acc0, false, false);
        acc1 = __builtin_amdgcn_wmma_f32_16x16x4_f32(false, a1, false, a1,
                                                     (short)0, acc1, false, false);
    }
    v8f G = acc0 + acc1;

    // ---- scatter accumulator to LDS: lane holds column r, rows hl*8+i ----
    #pragma unroll
    for (int i = 0; i < 8; ++i)
        gbuf[wave][(hl * 8 + i) * 16 + r] = G[i];
#if __has_builtin(__builtin_amdgcn_s_wait_dscnt)
    __builtin_amdgcn_s_wait_dscnt(0);
#endif
    __asm__ volatile("" ::: "memory");

    // ---- per-pentachoron scalar tail: lanes 0..2 of each wave ----
    if (lane < PPW) {
        const long long pent = pentBase + lane;
        if (pent < (long long)V) {
            const int b = lane * 5;
            float g[5][5];
            #pragma unroll
            for (int i = 0; i < 5; ++i)
                #pragma unroll
                for (int j = 0; j < 5; ++j)
                    g[i][j] = gbuf[wave][(b + i) * 16 + (b + j)];

            float sq[5];
            #pragma unroll
            for (int i = 0; i < 5; ++i) sq[i] = g[i][i];

            float d2[5][5];
            #pragma unroll
            for (int i = 0; i < 5; ++i)
                #pragma unroll
                for (int j = 0; j < 5; ++j)
                    d2[i][j] = fmaxf(sq[i] + sq[j] - 2.0f * g[i][j], 0.0f);

            // Cayley-Menger determinant (6x6, partial-pivot LU)
            float A[6][6];
            A[0][0] = 0.0f;
            #pragma unroll
            for (int j = 1; j < 6; ++j) { A[0][j] = 1.0f; A[j][0] = 1.0f; }
            #pragma unroll
            for (int i = 0; i < 5; ++i)
                #pragma unroll
                for (int j = 0; j < 5; ++j)
                    A[i + 1][j + 1] = d2[i][j];

            float det = 1.0f;
            for (int c = 0; c < 6; ++c) {
                int piv = c; float mx = fabsf(A[c][c]);
                for (int rr = c + 1; rr < 6; ++rr) {
                    float a = fabsf(A[rr][c]);
                    if (a > mx) { mx = a; piv = rr; }
                }
                if (piv != c) {
                    for (int cc = c; cc < 6; ++cc) {
                        float t = A[c][cc]; A[c][cc] = A[piv][cc]; A[piv][cc] = t;
                    }
                    det = -det;
                }
                float p = A[c][c];
                det *= p;
                if (p != 0.0f) {
                    float inv = 1.0f / p;
                    for (int rr = c + 1; rr < 6; ++rr) {
                        float f = A[rr][c] * inv;
                        for (int cc = c + 1; cc < 6; ++cc) A[rr][cc] -= f * A[c][cc];
                    }
                }
            }
            float vol_sq = fmaxf(-det / 9216.0f, 0.0f);
            float volume = sqrtf(vol_sq);

            // edge stats over 10 unique edges, unbiased std
            float e[10]; int idx = 0; float esum = 0.0f;
            #pragma unroll
            for (int i = 0; i < 5; ++i)
                #pragma unroll
                for (int j = 0; j < 5; ++j)
                    if (j > i) { float ev = sqrtf(d2[i][j]); e[idx++] = ev; esum += ev; }
            float mean_edge = esum * 0.1f;
            float evar = 0.0f;
            #pragma unroll
            for (int i = 0; i < 10; ++i) { float d = e[i] - mean_edge; evar += d * d; }
            float std_edge = sqrtf(evar / 9.0f);

            // spread: derived entirely from the Gram
            float rowsum[5]; float S = 0.0f;
            #pragma unroll
            for (int i = 0; i < 5; ++i) {
                float s = 0.0f;
                #pragma unroll
                for (int j = 0; j < 5; ++j) s += g[i][j];
                rowsum[i] = s; S += s;
            }
            float dc[5]; float dsum = 0.0f;
            #pragma unroll
            for (int i = 0; i < 5; ++i) {
                float v = fmaxf(sq[i] - 0.4f * rowsum[i] + 0.04f * S, 0.0f);
                dc[i] = sqrtf(v); dsum += dc[i];
            }
            float mdc = dsum * 0.2f;
            float dvar = 0.0f;
            #pragma unroll
            for (int i = 0; i < 5; ++i) { float d = dc[i] - mdc; dvar += d * d; }
            float spread = sqrtf(dvar / 4.0f);

            float seed = sigmoidf_(volume * 10.0f) * 0.4f
                       + sigmoidf_(std_edge / (mean_edge + 1e-6f)) * 0.3f
                       + sigmoidf_(spread) * 0.3f;

            float x = fminf(fmaxf(seed, 1e-6f), 1.0f - 1e-6f);
            float cant = 0.0f, factor = 0.5f;
            #pragma unroll
            for (int it = 0; it < 8; ++it) {
                float xs = x * 3.0f;
                int d = (int)xs;
                x = xs - (float)d;
                if (d == 2) cant += factor;
                factor *= 0.5f;
            }
            out[pent] = fminf(fmaxf(cant, 0.0f), 1.0f);
        }
    }
}

extern "C" void kernel_launch(void* const* d_in, const int* in_sizes, int n_in,
                              void* d_out, int out_size, void* d_ws, size_t ws_size,
                              hipStream_t stream) {
    (void)n_in; (void)d_ws; (void)ws_size; (void)out_size;
    const float* P = (const float*)d_in[0];
    float* out = (float*)d_out;
    const int V = in_sizes[0] / (5 * DIMK);          // 262144 for the reference
    const long long tiles = ((long long)V + PPW - 1) / PPW;
    const int blocks = (int)((tiles + WAVES - 1) / WAVES);
    pentachora_fingerprint_kernel<<<blocks, WAVES * 32, 0, stream>>>(P, out, V);
}